// CAM_Module_23321672417845
// MI455X (gfx1250) — compile-verified
//
#include <hip/hip_runtime.h>
#include <math.h>

typedef float v2f __attribute__((ext_vector_type(2)));
typedef float v4f __attribute__((ext_vector_type(4)));
typedef float v8f __attribute__((ext_vector_type(8)));

#define Bb 32
#define Cc 2048
#define K2 4096

// ---------------------------------------------------------------------------
// Kernel 1: x[32][2048] = concat(x1,x2) @ W^T + bias  via V_WMMA_F32_16X16X4_F32
// One wave (32 threads) per 16x16 output tile. K loop steps 4.
// A (16x4 f32): lanes 0-15 hold row M=lane, K={0,1}; lanes 16-31 row M=lane-16, K={2,3}.
// B (4x16 f32): VGPR v, lanes 0-15 = row K=v, N=lane; lanes 16-31 = row K=v+2, N=lane-16.
// C/D (16x16): VGPR v, lanes 0-15 -> M=v; lanes 16-31 -> M=v+8.
// ---------------------------------------------------------------------------
__global__ void __launch_bounds__(32)
gemm_x_kernel(const float* __restrict__ x1, const float* __restrict__ x2,
              const float* __restrict__ W, const float* __restrict__ bias,
              float* __restrict__ xbuf) {
    const int lane = threadIdx.x;
    const int col  = lane & 15;
    const int hi   = lane >> 4;          // 0 or 1
    const int n0   = blockIdx.x * 16;    // 128 tiles over N=2048
    const int m0   = blockIdx.y * 16;    // 2 tiles over M=32

    const float bv = bias[n0 + col];
    v8f c;
#pragma unroll
    for (int v = 0; v < 8; ++v) c[v] = bv;

    // W is (C, 2C): B[k][n] = W[n][k]
    const float* wrow = W + (size_t)(n0 + col) * K2 + 2 * hi;

    const float* srcs[2] = { x1, x2 };
#pragma unroll
    for (int s = 0; s < 2; ++s) {
        const float* arow = srcs[s] + (size_t)(m0 + col) * Cc + 2 * hi;
        const float* wk   = wrow + s * Cc;
#pragma unroll 8
        for (int k = 0; k < Cc; k += 4) {
            v2f a = *(const v2f*)(arow + k);   // A[m][k+2hi], A[m][k+1+2hi]
            v2f b = *(const v2f*)(wk + k);     // B[k+2hi][n], B[k+1+2hi][n]
            c = __builtin_amdgcn_wmma_f32_16x16x4_f32(
                    false, a, false, b, (short)0, c, false, false);
        }
    }

    float* orow = xbuf + (size_t)(m0 + hi * 8) * Cc + (n0 + col);
#pragma unroll
    for (int v = 0; v < 8; ++v) orow[(size_t)v * Cc] = c[v];
}

// ---------------------------------------------------------------------------
// Kernel 2: per-batch row max/min of x
// ---------------------------------------------------------------------------
__global__ void __launch_bounds__(256)
minmax_kernel(const float* __restrict__ xbuf,
              float* __restrict__ xmax, float* __restrict__ xmin) {
    __shared__ float smax[256], smin[256];
    const int b = blockIdx.x, t = threadIdx.x;
    float mx = -INFINITY, mn = INFINITY;
#pragma unroll
    for (int j = t; j < Cc; j += 256) {
        float v = xbuf[b * Cc + j];
        mx = fmaxf(mx, v);
        mn = fminf(mn, v);
    }
    smax[t] = mx; smin[t] = mn;
    __syncthreads();
    for (int off = 128; off > 0; off >>= 1) {
        if (t < off) {
            smax[t] = fmaxf(smax[t], smax[t + off]);
            smin[t] = fminf(smin[t], smin[t + off]);
        }
        __syncthreads();
    }
    if (t == 0) { xmax[b] = smax[0]; xmin[b] = smin[0]; }
}

// ---------------------------------------------------------------------------
// Kernel 3: fused softmax row + attention write (NT stores) + out1/out2.
// One wave32 per (b,i) row. att[i,j] = exp(n_i - x_i*x_j)/denom.
// Each lane holds 64 exp values (16 x float4) so exp is evaluated exactly once.
// ---------------------------------------------------------------------------
__global__ void __launch_bounds__(256)
attn_kernel(const float* __restrict__ x1, const float* __restrict__ x2,
            const float* __restrict__ xbuf, const float* __restrict__ xmax,
            const float* __restrict__ xmin, float* __restrict__ out) {
    const int lane = threadIdx.x & 31;
    const int row  = blockIdx.x * (blockDim.x >> 5) + (threadIdx.x >> 5); // b*2048+i
    const int b    = row >> 11;

    const float xi = xbuf[row];
    const float ni = (xi > 0.0f) ? xi * xmin[b] : xi * xmax[b]; // min_j(x_i*x_j)

    const float* xr  = xbuf + (size_t)b * Cc;
    const float* x1r = x1   + (size_t)b * Cc;
    const float* x2r = x2   + (size_t)b * Cc;

    v4f ev[16];
    float s0 = 0.0f, s1 = 0.0f, s2 = 0.0f;
#pragma unroll
    for (int c = 0; c < 16; ++c) {
        const int j = c * 128 + lane * 4;
        v4f xv  = *(const v4f*)(xr  + j);
        v4f x1v = *(const v4f*)(x1r + j);
        v4f x2v = *(const v4f*)(x2r + j);
        v4f e;
        e.x = __expf(fmaf(-xi, xv.x, ni));
        e.y = __expf(fmaf(-xi, xv.y, ni));
        e.z = __expf(fmaf(-xi, xv.z, ni));
        e.w = __expf(fmaf(-xi, xv.w, ni));
        ev[c] = e;
        s0 += (e.x + e.y) + (e.z + e.w);
        s1 += e.x * x1v.x + e.y * x1v.y + e.z * x1v.z + e.w * x1v.w;
        s2 += e.x * x2v.x + e.y * x2v.y + e.z * x2v.z + e.w * x2v.w;
    }

    // wave32 all-reduce
#pragma unroll
    for (int off = 16; off > 0; off >>= 1) {
        s0 += __shfl_xor(s0, off, 32);
        s1 += __shfl_xor(s1, off, 32);
        s2 += __shfl_xor(s2, off, 32);
    }
    const float inv = 1.0f / s0;   // s0 >= 1 (one term is exp(0))

    float* att = out + 2 * (Bb * Cc) + (size_t)row * Cc;
#pragma unroll
    for (int c = 0; c < 16; ++c) {
        const int j = c * 128 + lane * 4;
        v4f e = ev[c];
        e.x *= inv; e.y *= inv; e.z *= inv; e.w *= inv;
        __builtin_nontemporal_store(e, (v4f*)(att + j));  // 537MB stream, don't cache
    }

    if (lane == 0) {
        out[row]           = fmaf(s1, inv, x1[row]);  // out1 = att@x1 + x1
        out[Bb * Cc + row] = fmaf(s2, inv, x2[row]);  // out2 = att@x2 + x2
    }
}

// ---------------------------------------------------------------------------
extern "C" void kernel_launch(void* const* d_in, const int* in_sizes, int n_in,
                              void* d_out, int out_size, void* d_ws, size_t ws_size,
                              hipStream_t stream) {
    const float* x1   = (const float*)d_in[0];
    const float* x2   = (const float*)d_in[1];
    const float* W    = (const float*)d_in[2];
    const float* bias = (const float*)d_in[3];
    float* out  = (float*)d_out;

    float* xbuf = (float*)d_ws;          // 32*2048 floats
    float* xmax = xbuf + Bb * Cc;        // 32 floats
    float* xmin = xmax + Bb;             // 32 floats

    gemm_x_kernel<<<dim3(Cc / 16, Bb / 16), 32, 0, stream>>>(x1, x2, W, bias, xbuf);
    minmax_kernel<<<Bb, 256, 0, stream>>>(xbuf, xmax, xmin);
    attn_kernel<<<(Bb * Cc) / 8, 256, 0, stream>>>(x1, x2, xbuf, xmax, xmin, out);
}